// F_29987461661363
// MI455X (gfx1250) — compile-verified
//
#include <hip/hip_runtime.h>

// ---------------------------------------------------------------------------
// Incompressible NS projection step, 512x512 periodic grid, fp32.
// ws layout (floats): partial[2*N] | B[N] | phiA[N] | phiB[N]  => 5 MB needed.
// Dominant phase: 1000 Jacobi sweeps, entirely L2-resident on MI455X (192MB L2).
// Each wave owns a 16x32 strip (two 16x16 tiles): nsum = S*P + P*S + halo,
// computed with 16x v_wmma_f32_16x16x4_f32 (exact: S is a 0/1 shift matrix).
// RHS strip staged to LDS via the Tensor Data Mover (tensor_load_to_lds +
// s_wait_tensorcnt). All halo handling is branchless (EXEC stays all-1s).
// ---------------------------------------------------------------------------

#define NGRID 512
#define NPTS  (NGRID * NGRID)      // 262144 = 2^18
#define NU_   0.1f
#define RHODT 10.0f                // RHO/DT
#define STEPS 1000

typedef __attribute__((ext_vector_type(2))) float        v2f;
typedef __attribute__((ext_vector_type(8))) float        v8f;
typedef __attribute__((ext_vector_type(4))) unsigned int v4u;
typedef __attribute__((ext_vector_type(8))) int          v8i;
typedef __attribute__((ext_vector_type(4))) int          v4i;

__device__ __forceinline__ int wrap(int i) { return i & (NGRID - 1); }

// -------------------- phase 1: partial = -adv + NU*lap + forcing -----------
__global__ __launch_bounds__(256) void partial_kernel(const float* __restrict__ X,
                                                      const float* __restrict__ F,
                                                      float* __restrict__ partial) {
    int idx = blockIdx.x * 256 + threadIdx.x;     // 0 .. 2*NPTS-1
    int c   = idx >> 18;
    int p   = idx & (NPTS - 1);
    int x = p >> 9, y = p & 511;
    const float* f = X + c * NPTS;
    const float* u = X;                // X[0]
    const float* v = X + NPTS;         // X[1]
    int xm = wrap(x - 1), xp = wrap(x + 1), ym = wrap(y - 1), yp = wrap(y + 1);
    float fxm = f[xm * NGRID + y], fxp = f[xp * NGRID + y];
    float fym = f[x * NGRID + ym], fyp = f[x * NGRID + yp];
    float fc  = f[p];
    float dfdx = 0.5f * (fxp - fxm);
    float dfdy = 0.5f * (fyp - fym);
    float lap  = fxm + fxp + fym + fyp - 4.0f * fc;   // DX = 1
    float adv  = u[p] * dfdx + v[p] * dfdy;
    partial[idx] = -adv + NU_ * lap + F[idx];
}

// -------------------- phase 2: B = RHO/DT * div(partial); phi0 = B ---------
__global__ __launch_bounds__(256) void rhs_kernel(const float* __restrict__ partial,
                                                  float* __restrict__ B,
                                                  float* __restrict__ phi0) {
    int p = blockIdx.x * 256 + threadIdx.x;
    int x = p >> 9, y = p & 511;
    const float* p0 = partial;
    const float* p1 = partial + NPTS;
    float dpx = 0.5f * (p0[wrap(x + 1) * NGRID + y] - p0[wrap(x - 1) * NGRID + y]);
    float dpy = 0.5f * (p1[x * NGRID + wrap(y + 1)] - p1[x * NGRID + wrap(y - 1)]);
    float b = RHODT * (dpx + dpy);
    B[p] = b;
    phi0[p] = b;
}

// -------------------- phase 3: one Jacobi sweep (WMMA + TDM) ---------------
__device__ __forceinline__ v8f wmma4(v2f a, v2f b, v8f c) {
    // D = A(16x4,f32) x B(4x16,f32) + C(16x16,f32)
    return __builtin_amdgcn_wmma_f32_16x16x4_f32(false, a, false, b, (short)0, c,
                                                 false, false);
}

__global__ __launch_bounds__(256) void jacobi_kernel(const float* __restrict__ phiIn,
                                                     const float* __restrict__ Brhs,
                                                     float* __restrict__ phiOut) {
    __shared__ float smemB[8 * 512];                 // one 16x32 strip per wave

    int lane = (int)(threadIdx.x & 31);
    int wave = __builtin_amdgcn_readfirstlane((int)(threadIdx.x >> 5));
    int strip = blockIdx.x * 8 + wave;               // 512 strips, 8 per block
    int ti = strip >> 4, tjp = strip & 15;           // 32 strip-rows x 16 strip-cols
    int row0  = ti * 16;
    int col0L = tjp * 32;
    int col0R = col0L + 16;
    int half = lane >> 4;                            // 0: lanes 0-15, 1: 16-31
    int lc   = lane & 15;

    // ---- TDM: async-load this wave's 16x32 RHS strip into LDS ----
    {
        unsigned lds_off = (unsigned)(size_t)(&smemB[0]) + (unsigned)(wave * 2048);
        unsigned long long ga =
            (unsigned long long)(size_t)(Brhs + (size_t)row0 * NGRID + col0L);
        v4u g0 = { 1u,                                    // count=1, user descriptor
                   lds_off,                               // lds_addr
                   (unsigned)(ga & 0xffffffffu),          // global_addr[31:0]
                   (unsigned)((ga >> 32) & 0x01ffffffull) // global_addr[56:32]
                       | 0x80000000u };                   // type=2 ("image")
        v8i g1 = { (int)(2u << 16),     // data_size = 4 bytes
                   (int)(512u << 16),   // tensor_dim0 = 512 (bits 63:48)
                   (int)(512u << 16),   // tensor_dim1 = 512 (bits 95:80)
                   (int)(32u << 16),    // tile_dim0 = 32   (bits 127:112)
                   16,                  // tile_dim1 = 16   (bits 143:128)
                   512,                 // tensor_dim0_stride (bits 207:160, low 32)
                   0, 0 };
        v4i z4 = { 0, 0, 0, 0 };
        v8i z8 = { 0, 0, 0, 0, 0, 0, 0, 0 };
        __builtin_amdgcn_tensor_load_to_lds(g0, g1, z4, z4, z8, 0);
    }

    // ---- shift matrix S slices via per-lane neighbor bitmask (branchless) ----
    // mask bit c == 1  <=>  |lc - c| == 1
    unsigned nmask = (2u << lc) | ((1u << lc) >> 1);
    v2f sk[4];
#pragma unroll
    for (int k = 0; k < 4; ++k) {
        int c0 = 4 * k + 2 * half;
        sk[k][0] = ((nmask >> c0) & 1u) ? 1.0f : 0.0f;
        sk[k][1] = ((nmask >> (c0 + 1)) & 1u) ? 1.0f : 0.0f;
    }

    // ---- load both P tiles in WMMA A-layout (float2) and B-layout ----
    v2f paL[4], pbL[4], paR[4], pbR[4];
#pragma unroll
    for (int k = 0; k < 4; ++k) {
        int c0 = 4 * k + 2 * half;
        const float* rowp = phiIn + (size_t)(row0 + lc) * NGRID;
        paL[k] = *(const v2f*)(rowp + col0L + c0);
        paR[k] = *(const v2f*)(rowp + col0R + c0);
        const float* kr0 = phiIn + (size_t)(row0 + c0)     * NGRID;
        const float* kr1 = phiIn + (size_t)(row0 + c0 + 1) * NGRID;
        pbL[k][0] = kr0[col0L + lc];
        pbL[k][1] = kr1[col0L + lc];
        pbR[k][0] = kr0[col0R + lc];
        pbR[k][1] = kr1[col0R + lc];
    }

    // ---- nsum_interior = S*P + P*S via 2 x 8 chained 16x16x4 f32 WMMAs ----
    v8f accL = {}, accR = {};
#pragma unroll
    for (int k = 0; k < 4; ++k) accL = wmma4(sk[k], pbL[k], accL);  // S*P (x)
#pragma unroll
    for (int k = 0; k < 4; ++k) accL = wmma4(paL[k], sk[k], accL);  // P*S (y)
#pragma unroll
    for (int k = 0; k < 4; ++k) accR = wmma4(sk[k], pbR[k], accR);
#pragma unroll
    for (int k = 0; k < 4; ++k) accR = wmma4(paR[k], sk[k], accR);

    // ---- halo rows: lanes 0-15 fetch the row above, lanes 16-31 row below ----
    {
        int hrow = wrap(row0 - 1 + half * 17);       // half=0: row0-1, half=1: row0+16
        float hrL = phiIn[(size_t)hrow * NGRID + col0L + lc];
        float hrR = phiIn[(size_t)hrow * NGRID + col0R + lc];
        accL[0] += (half == 0) ? hrL : 0.0f;         // C/D row 0: reg0, lanes 0-15
        accL[7] += (half == 1) ? hrL : 0.0f;         // C/D row 15: reg7, lanes 16-31
        accR[0] += (half == 0) ? hrR : 0.0f;
        accR[7] += (half == 1) ? hrR : 0.0f;
    }

    // ---- halo columns (branchless address, masked add) ----
    int rb  = row0 + (half ? 8 : 0);
    int off = (lc == 15) - (lc == 0);                // -1 at lc==0, +1 at lc==15
    float cm = (off != 0) ? 1.0f : 0.0f;
    int scolL = wrap(col0L + lc + off);
    int scolR = wrap(col0R + lc + off);
#pragma unroll
    for (int r = 0; r < 8; ++r) {
        const float* rowp = phiIn + (size_t)(rb + r) * NGRID;
        accL[r] += cm * rowp[scolL];
        accR[r] += cm * rowp[scolR];
    }

    // ---- wait for TDM, apply Jacobi update, store in C/D layout ----
    __builtin_amdgcn_s_wait_tensorcnt(0);
    int sbase = wave * 512 + (half ? 8 : 0) * 32 + lc;
#pragma unroll
    for (int r = 0; r < 8; ++r) {
        float bvL = smemB[sbase + r * 32];
        float bvR = smemB[sbase + r * 32 + 16];
        float* rowp = phiOut + (size_t)(rb + r) * NGRID;
        rowp[col0L + lc] = (accL[r] - bvL) * 0.25f;
        rowp[col0R + lc] = (accR[r] - bvR) * 0.25f;
    }
}

// -------------------- phase 4: out = partial - grad(phi) (RHO = 1) ---------
__global__ __launch_bounds__(256) void final_kernel(const float* __restrict__ partial,
                                                    const float* __restrict__ phi,
                                                    float* __restrict__ out) {
    int idx = blockIdx.x * 256 + threadIdx.x;   // 0 .. 2*NPTS-1
    int c   = idx >> 18;
    int p   = idx & (NPTS - 1);
    int x = p >> 9, y = p & 511;
    float g;
    if (c == 0)
        g = 0.5f * (phi[wrap(x + 1) * NGRID + y] - phi[wrap(x - 1) * NGRID + y]);
    else
        g = 0.5f * (phi[x * NGRID + wrap(y + 1)] - phi[x * NGRID + wrap(y - 1)]);
    out[idx] = partial[idx] - g;
}

// ---------------------------------------------------------------------------
extern "C" void kernel_launch(void* const* d_in, const int* in_sizes, int n_in,
                              void* d_out, int out_size, void* d_ws, size_t ws_size,
                              hipStream_t stream) {
    // d_in[0] = t (scalar, unused), d_in[1] = X (2*N), d_in[2] = forcing (2*N)
    const float* X = (const float*)d_in[1];
    const float* F = (const float*)d_in[2];
    float* ws      = (float*)d_ws;
    float* partial = ws;                 // 2*NPTS
    float* B       = ws + 2 * NPTS;      // NPTS
    float* phiA    = ws + 3 * NPTS;      // NPTS
    float* phiB    = ws + 4 * NPTS;      // NPTS
    float* out     = (float*)d_out;      // 2*NPTS

    partial_kernel<<<2 * NPTS / 256, 256, 0, stream>>>(X, F, partial);
    rhs_kernel<<<NPTS / 256, 256, 0, stream>>>(partial, B, phiA);

    for (int i = 0; i < STEPS; ++i) {
        const float* src = (i & 1) ? phiB : phiA;
        float*       dst = (i & 1) ? phiA : phiB;
        jacobi_kernel<<<64, 256, 0, stream>>>(src, B, dst);
    }
    // STEPS is even -> final phi in phiA
    final_kernel<<<2 * NPTS / 256, 256, 0, stream>>>(partial, phiA, out);
}